// GNNModel_58643483459879
// MI455X (gfx1250) — compile-verified
//
#include <hip/hip_runtime.h>
#include <hip/hip_bf16.h>

typedef __attribute__((ext_vector_type(2))) float v2f;
typedef __attribute__((ext_vector_type(8))) float v8f;

// ---------------------------------------------------------------------------
// Tall-skinny GEMM: out[s] (+)= A[M x K] @ B[K x 16] over k-slice s=blockIdx.y
//   One wave computes one 16x16 output tile using V_WMMA_F32_16X16X4_F32.
//   A-operand (16x4 f32): lanes 0-15 -> rows 0-15 K={k,k+1} (2 VGPRs),
//                         lanes 16-31 -> rows 0-15 K={k+2,k+3}.
//   B-operand (4x16 f32): lanes 0-15 -> cols, K={k,k+1}; lanes 16-31 K={k+2,k+3}.
//   C/D (16x16 f32): vgpr j, lanes 0-15 -> M=j; lanes 16-31 -> M=j+8.
// Main loop is software-pipelined by 8 WMMA steps (K += 32): 24 loads issued
// as a batch, then 8 WMMAs, so the compiler can stage s_wait_loadcnt instead
// of a full wait before every WMMA.
// ---------------------------------------------------------------------------
__global__ __launch_bounds__(128) void gcn_wmma_gemm_n16(
    const float* __restrict__ A, const float* __restrict__ B,
    float* __restrict__ out, int M, int lda, int klen) {
  const int lane   = threadIdx.x & 31;
  const int wave   = threadIdx.x >> 5;
  const int rowblk = blockIdx.x * (blockDim.x >> 5) + wave;
  const int g      = lane >> 4;    // K half-select (0: K+0/1, 1: K+2/3)
  const int c      = lane & 15;    // row (for A) / col (for B)

  const float* ap = A + (size_t)(rowblk * 16 + c) * (size_t)lda + (g * 2);
  const float* bp = B + g * 32 + c;   // row (g*2) of B, column c

  const int k0     = blockIdx.y * klen;
  const int kmain  = k0 + (klen & ~31);   // unrolled-by-32 portion
  const int kend   = k0 + klen;

  v8f acc = {};

  for (int k = k0; k < kmain; k += 32) {
    // ---- load batch for 8 WMMA steps (1x b64 + 2x b32 per step) ----
    v2f a[8], b[8];
#pragma unroll
    for (int u = 0; u < 8; ++u) {
      const int kk = k + 4 * u;
      a[u] = *(const v2f*)(ap + kk);              // global_load_b64
      b[u].x = bp[(size_t)kk * 16];               // global_load_b32
      b[u].y = bp[(size_t)kk * 16 + 16];          // global_load_b32
    }
    // one line per A-row is consumed per body; prefetch the next stretch
    __builtin_prefetch(ap + k + 512, 0, 3);
    // ---- 8 dependent accumulating WMMAs ----
#pragma unroll
    for (int u = 0; u < 8; ++u) {
      acc = __builtin_amdgcn_wmma_f32_16x16x4_f32(
          false, a[u], false, b[u], (short)0, acc, false, false);
    }
  }

  // remainder (handles klen==16 for the H1@W2 GEMM; skipped when klen%32==0)
  for (int k = kmain; k < kend; k += 4) {
    v2f a = *(const v2f*)(ap + k);
    v2f b;
    b.x = bp[(size_t)k * 16];
    b.y = bp[(size_t)k * 16 + 16];
    acc = __builtin_amdgcn_wmma_f32_16x16x4_f32(
        false, a, false, b, (short)0, acc, false, false);
  }

  // store 16x16 tile: vgpr j -> row rowblk*16 + j + 8*g, col c
  float* op = out + (size_t)blockIdx.y * (size_t)M * 16 +
              ((size_t)rowblk * 16 + 8 * g) * 16 + c;
#pragma unroll
  for (int j = 0; j < 8; ++j) op[j * 16] = acc[j];
}

// ---------------------------------------------------------------------------
// Sum 4 k-slice partials, add per-column bias, optional ReLU.
// ---------------------------------------------------------------------------
__global__ __launch_bounds__(256) void gcn_reduce4_bias(
    const float* __restrict__ parts, size_t pstride,
    const float* __restrict__ bias, float* __restrict__ out,
    int total, int do_relu) {
  int i = blockIdx.x * blockDim.x + threadIdx.x;
  if (i >= total) return;
  float v = parts[i] + parts[i + pstride] + parts[i + 2 * pstride] +
            parts[i + 3 * pstride];
  v += bias[i & 15];
  if (do_relu) v = fmaxf(v, 0.0f);
  out[i] = v;
}

// ---------------------------------------------------------------------------
// Per-row: softmax over 16 classes then dot with Wd -> out[row]
// ---------------------------------------------------------------------------
__global__ __launch_bounds__(256) void gcn_softmax_head(
    const float* __restrict__ S, const float* __restrict__ Wd,
    const float* __restrict__ bd, float* __restrict__ out, int M) {
  int r = blockIdx.x * blockDim.x + threadIdx.x;
  if (r >= M) return;
  const float* s = S + (size_t)r * 16;
  float v[16];
  float mx = -3.402823466e+38f;
#pragma unroll
  for (int n = 0; n < 16; ++n) { v[n] = s[n]; mx = fmaxf(mx, v[n]); }
  float z = 0.0f, acc = 0.0f;
#pragma unroll
  for (int n = 0; n < 16; ++n) {
    float e = __expf(v[n] - mx);
    z += e;
    acc += e * Wd[n];
  }
  out[r] = acc / z + bd[0];
}

extern "C" void kernel_launch(void* const* d_in, const int* in_sizes, int n_in,
                              void* d_out, int out_size, void* d_ws, size_t ws_size,
                              hipStream_t stream) {
  const float* x   = (const float*)d_in[0];  // [N,F]
  const float* adj = (const float*)d_in[1];  // [N,N]
  const float* W1  = (const float*)d_in[2];  // [F,16]
  const float* b1  = (const float*)d_in[3];  // [16]
  const float* W2  = (const float*)d_in[4];  // [16,16]
  const float* b2  = (const float*)d_in[5];  // [16]
  const float* Wd  = (const float*)d_in[6];  // [16,1]
  const float* bd  = (const float*)d_in[7];  // [1]
  float* out = (float*)d_out;                // [N,1]

  const int N = 16384, F = 512;
  float* ws   = (float*)d_ws;
  float* P1   = ws;                    // N*16
  float* H1   = ws + (size_t)N * 16;   // N*16
  float* P2   = ws + (size_t)N * 32;   // N*16
  float* S    = ws + (size_t)N * 48;   // N*16
  float* PART = ws + (size_t)N * 64;   // 4 * N*16

  dim3 blk(128);                       // 4 waves/block, 4 row-tiles/block

  // 1) P1 = X @ W1                       (K=512, no split)
  gcn_wmma_gemm_n16<<<dim3(N / 64, 1), blk, 0, stream>>>(x, W1, P1, N, F, F);
  // 2) PART[0..3] = A @ P1 (4-way K-split over the 1 GiB adjacency stream)
  gcn_wmma_gemm_n16<<<dim3(N / 64, 4), blk, 0, stream>>>(adj, P1, PART, N, N, N / 4);
  // 3) H1 = relu(sum(PART) + b1)
  gcn_reduce4_bias<<<(N * 16) / 256, 256, 0, stream>>>(PART, (size_t)N * 16, b1, H1,
                                                       N * 16, 1);
  // 4) P2 = H1 @ W2                      (K=16, remainder loop only)
  gcn_wmma_gemm_n16<<<dim3(N / 64, 1), blk, 0, stream>>>(H1, W2, P2, N, 16, 16);
  // 5) PART[0..3] = A @ P2 (second adjacency pass)
  gcn_wmma_gemm_n16<<<dim3(N / 64, 4), blk, 0, stream>>>(adj, P2, PART, N, N, N / 4);
  // 6) S = sum(PART) + b2
  gcn_reduce4_bias<<<(N * 16) / 256, 256, 0, stream>>>(PART, (size_t)N * 16, b2, S,
                                                       N * 16, 0);
  // 7) out = softmax(S) @ Wd + bd
  gcn_softmax_head<<<N / 256, 256, 0, stream>>>(S, Wd, bd, out, N);
}